// V1Column_7670811591210
// MI455X (gfx1250) — compile-verified
//
#include <hip/hip_runtime.h>
#include <cstdint>

#define NB 5
#define ZWORDS_MAX 3136   // 100352 neuron bits; covers N_NEURONS=100000
#define BASIS_MAX  2560   // 512 types * 5 basis

#if defined(__AMDGCN__) && defined(__has_builtin)
# if __has_builtin(__builtin_amdgcn_global_load_async_to_lds_b32)
#  define HAVE_ASYNC_LDS 1
# endif
# if __has_builtin(__builtin_amdgcn_global_load_async_to_lds_b128)
#  define HAVE_ASYNC_LDS_B128 1
# endif
#endif

typedef int v4i __attribute__((ext_vector_type(4)));

typedef __attribute__((address_space(1))) int  gint_t;   // global (AS1) int
typedef __attribute__((address_space(3))) int  lint_t;   // LDS (AS3) int
typedef __attribute__((address_space(1))) v4i  gv4i_t;   // global (AS1) int4
typedef __attribute__((address_space(3))) v4i  lv4i_t;   // LDS (AS3) int4

__device__ __forceinline__ gint_t* as_global_i32(const void* p) {
  return (gint_t*)(uintptr_t)p;
}
__device__ __forceinline__ lint_t* as_lds_i32(const void* p) {
  // generic LDS pointer: addr[31:0] is the DS offset; AS3 pointers are 32-bit
  return (lint_t*)(uint32_t)(uintptr_t)p;
}
__device__ __forceinline__ gv4i_t* as_global_v4i(const void* p) {
  return (gv4i_t*)(uintptr_t)p;
}
__device__ __forceinline__ lv4i_t* as_lds_v4i(const void* p) {
  return (lv4i_t*)(uint32_t)(uintptr_t)p;
}

__device__ __forceinline__ void async_wait0() {
#if defined(HAVE_ASYNC_LDS)
# if __has_builtin(__builtin_amdgcn_s_wait_asynccnt)
  __builtin_amdgcn_s_wait_asynccnt(0);
# else
  asm volatile("s_wait_asynccnt 0" ::: "memory");
# endif
#endif
}

// Copy nwords 32-bit words global -> LDS, cooperatively across the block.
// Uses async b128 DMA for the bulk, async b32 for the tail.
__device__ __forceinline__ void stage_words(const uint32_t* __restrict__ gsrc,
                                            uint32_t* __restrict__ ldst,
                                            int nwords, int tid, int nthreads) {
#if defined(HAVE_ASYNC_LDS)
  const int nvec = nwords >> 2;   // 16B chunks
# if defined(HAVE_ASYNC_LDS_B128)
  for (int i = tid; i < nvec; i += nthreads) {
    __builtin_amdgcn_global_load_async_to_lds_b128(as_global_v4i(gsrc + 4 * i),
                                                   as_lds_v4i(ldst + 4 * i),
                                                   /*offset=*/0, /*cpol=*/0);
  }
  for (int i = (nvec << 2) + tid; i < nwords; i += nthreads) {
    __builtin_amdgcn_global_load_async_to_lds_b32(as_global_i32(gsrc + i),
                                                  as_lds_i32(ldst + i), 0, 0);
  }
# else
  for (int i = tid; i < nwords; i += nthreads) {
    __builtin_amdgcn_global_load_async_to_lds_b32(as_global_i32(gsrc + i),
                                                  as_lds_i32(ldst + i), 0, 0);
  }
# endif
#else
  for (int i = tid; i < nwords; i += nthreads) ldst[i] = gsrc[i];
#endif
}

// ---------------------------------------------------------------------------
// Kernel 1: zero output, compress spike vector (f32 0/1) into a bitmask.
// ---------------------------------------------------------------------------
__global__ __launch_bounds__(256) void v1_init(const float* __restrict__ z,
                                               uint32_t* __restrict__ zbits,
                                               float* __restrict__ out,
                                               int n_neurons, int n_zwords,
                                               int out_elems) {
  const int t = blockIdx.x * blockDim.x + threadIdx.x;
  const int stride = gridDim.x * blockDim.x;
  const float4* z4 = (const float4*)z;
  for (int i = t; i < n_zwords; i += stride) {
    uint32_t m = 0;
    const int base = i * 32;
    if (base + 32 <= n_neurons) {
      // fast path: 8 x b128 loads, 128B per word
#pragma unroll
      for (int q = 0; q < 8; ++q) {
        const float4 v = z4[(base >> 2) + q];
        m |= (v.x > 0.0f) ? (1u << (4 * q + 0)) : 0u;
        m |= (v.y > 0.0f) ? (1u << (4 * q + 1)) : 0u;
        m |= (v.z > 0.0f) ? (1u << (4 * q + 2)) : 0u;
        m |= (v.w > 0.0f) ? (1u << (4 * q + 3)) : 0u;
      }
    } else {
      const int lim = n_neurons - base;
      for (int j = 0; j < lim; ++j)
        m |= (z[base + j] > 0.0f) ? (1u << j) : 0u;
    }
    zbits[i] = m;
  }
  // out_elems = n_post*NB = 500000, multiple of 4: zero with b128 stores
  float4* o4 = (float4*)out;
  const int n4 = out_elems >> 2;
  const float4 zf4 = make_float4(0.f, 0.f, 0.f, 0.f);
  for (int i = t; i < n4; i += stride) o4[i] = zf4;
  for (int i = (n4 << 2) + t; i < out_elems; i += stride) out[i] = 0.0f;
}

// ---------------------------------------------------------------------------
// Kernel 2: persistent grid-stride scatter.
//   LDS: spike bitmask (12.5 KB) + basis table (10 KB), staged via async DMA.
//   Stream: 4 synapses/thread with b128 loads; ~1% fire -> few f32 atomics.
// ---------------------------------------------------------------------------
__global__ __launch_bounds__(512) void v1_scatter(
    const int*   __restrict__ syn_idx,   // (n_syn, 2): post, pre
    const float* __restrict__ w,         // (n_syn,)
    const int*   __restrict__ syn_ids,   // (n_syn,)
    const float* __restrict__ basis_g,   // (n_types, NB)
    const uint32_t* __restrict__ zbits_g,
    float* __restrict__ out,             // (n_post, NB)
    int n_syn, int n_zwords, int n_basis_elems) {
  __shared__ __align__(16) uint32_t zbits[ZWORDS_MAX];
  __shared__ __align__(16) float    sbasis[BASIS_MAX];

  const int tid = threadIdx.x;
  const int nth = blockDim.x;

  const int zw = (n_zwords < ZWORDS_MAX) ? n_zwords : ZWORDS_MAX;
  const int bw = (n_basis_elems < BASIS_MAX) ? n_basis_elems : BASIS_MAX;
  stage_words(zbits_g, zbits, zw, tid, nth);
  stage_words((const uint32_t*)basis_g, (uint32_t*)sbasis, bw, tid, nth);
  async_wait0();
  __syncthreads();

  const bool basis_in_lds = (n_basis_elems <= BASIS_MAX);
  const bool z_in_lds     = (n_zwords     <= ZWORDS_MAX);

  auto process = [&](int post, int pre, float wgt, int stype) {
    const unsigned widx = ((unsigned)pre) >> 5;
    const uint32_t word = z_in_lds ? zbits[widx] : zbits_g[widx];
    if ((word >> (pre & 31)) & 1u) {
      float* o = out + (size_t)post * NB;
      const int b = stype * NB;
      if (basis_in_lds) {
#pragma unroll
        for (int r = 0; r < NB; ++r)
          unsafeAtomicAdd(&o[r], wgt * sbasis[b + r]);
      } else {
#pragma unroll
        for (int r = 0; r < NB; ++r)
          unsafeAtomicAdd(&o[r], wgt * basis_g[b + r]);
      }
    }
  };

  const int4*   idx4 = (const int4*)syn_idx;
  const float4* w4   = (const float4*)w;
  const int4*   id4  = (const int4*)syn_ids;
  const int ngroups = n_syn >> 2;
  const int gtid    = blockIdx.x * nth + tid;
  const int gstride = gridDim.x * nth;

  for (int g = gtid; g < ngroups; g += gstride) {
    // prefetch next stride's lines into cache (global_prefetch_b8)
    if (g + gstride < ngroups) {
      __builtin_prefetch(&idx4[2 * (g + gstride)], 0, 1);
      __builtin_prefetch(&w4[g + gstride], 0, 1);
      __builtin_prefetch(&id4[g + gstride], 0, 1);
    }
    const int4   pp0 = idx4[2 * g];      // post0 pre0 post1 pre1
    const int4   pp1 = idx4[2 * g + 1];  // post2 pre2 post3 pre3
    const float4 wv  = w4[g];
    const int4   tv  = id4[g];
    process(pp0.x, pp0.y, wv.x, tv.x);
    process(pp0.z, pp0.w, wv.y, tv.y);
    process(pp1.x, pp1.y, wv.z, tv.z);
    process(pp1.z, pp1.w, wv.w, tv.w);
  }
  // scalar tail (n_syn not a multiple of 4)
  for (int s = (ngroups << 2) + gtid; s < n_syn; s += gstride)
    process(syn_idx[2 * s], syn_idx[2 * s + 1], w[s], syn_ids[s]);
}

extern "C" void kernel_launch(void* const* d_in, const int* in_sizes, int n_in,
                              void* d_out, int out_size, void* d_ws, size_t ws_size,
                              hipStream_t stream) {
  const float*    z       = (const float*)d_in[0];  // (1, N_NEURONS)
  const int*      syn_idx = (const int*)d_in[1];    // (N_SYN, 2)
  const float*    w       = (const float*)d_in[2];  // (N_SYN,)
  const int*      sid     = (const int*)d_in[3];    // (N_SYN,)
  const float*    basis   = (const float*)d_in[4];  // (512, NB)

  const int n_neurons = in_sizes[0];          // BATCH == 1
  const int n_syn     = in_sizes[2];
  const int n_basis   = in_sizes[4];
  const int n_zwords  = (n_neurons + 31) / 32;

  uint32_t* zbits = (uint32_t*)d_ws;
  float*    out   = (float*)d_out;

  v1_init<<<1024, 256, 0, stream>>>(z, zbits, out, n_neurons, n_zwords, out_size);
  v1_scatter<<<1024, 512, 0, stream>>>(syn_idx, w, sid, basis, zbits, out,
                                       n_syn, n_zwords, n_basis);
}